// GAT_GCN_no_p_72232759984818
// MI455X (gfx1250) — compile-verified
//
#include <hip/hip_runtime.h>
#include <math.h>

// ---------------- CDNA5 WMMA types ----------------
typedef float v2f __attribute__((ext_vector_type(2)));
typedef float v8f __attribute__((ext_vector_type(8)));

#define F_IN   78
#define F_MID  156
#define F_OUT  312   /* HEADS * F_MID */
#define NB     128   /* batch graphs */

__device__ __forceinline__ float lrelu(float x, float s) { return x > 0.f ? x : x * s; }

// Order-preserving f32 -> u32 map (exact float max via integer atomicMax)
__device__ __forceinline__ unsigned fmap_ord(float x) {
  unsigned b = __float_as_uint(x);
  return (b & 0x80000000u) ? ~b : (b | 0x80000000u);
}
__device__ __forceinline__ float funmap_ord(unsigned u) {
  return __uint_as_float((u & 0x80000000u) ? (u & 0x7FFFFFFFu) : ~u);
}
#define NEG_INF_ORD 0x007FFFFFu   /* fmap_ord(-inf) */

// ---------------- generic fp32 WMMA GEMM: C = act(A[MxK] @ B[KxN] + bias) ----
// wave = one 16x16 C tile, K stepped by 4 via V_WMMA_F32_16X16X4_F32.
// Requires K even (all call sites satisfy this) so A lane-pairs are 8B aligned.
// Edge handling: row/col indices are CLAMPED (not predicated) -> unconditional
// loads, garbage only lands in never-stored output positions. Only the K tail
// (K % 4 != 0) is masked, via cndmask selects on clamped loads.
// act: 0 = none, 1 = leaky(0.01), 2 = relu
__global__ __launch_bounds__(256) void wmma_gemm_f32(
    const float* __restrict__ A, const float* __restrict__ B,
    const float* __restrict__ bias, float* __restrict__ C,
    int M, int K, int N, int act) {
  const int lane = threadIdx.x & 31;
  const int wave = blockIdx.x * (blockDim.x >> 5) + (threadIdx.x >> 5);
  const int tilesN = (N + 15) >> 4;
  const int tilesM = (M + 15) >> 4;
  if (wave >= tilesM * tilesN) return;          // wave-uniform exit: EXEC stays all-1s
  const int tm = wave / tilesN, tn = wave % tilesN;
  const int row0 = tm << 4, col0 = tn << 4;
  const int half = lane >> 4;                   // 0: lanes 0-15, 1: lanes 16-31
  int mrow = row0 + (lane & 15);                // A row for this lane
  int ncol = col0 + (lane & 15);                // B/C column for this lane
  const bool nvalid = ncol < N;
  const int mrow_c = mrow < M ? mrow : M - 1;   // clamped (always valid address)
  const int ncol_c = ncol < N ? ncol : N - 1;

  const float* __restrict__ Arow = A + (size_t)mrow_c * K;   // even K => 8B aligned pairs
  const float* __restrict__ Bcol = B + ncol_c;

  v8f acc = {};
  const int K4 = K & ~3;
  int k0 = 0;
#pragma unroll 4
  for (; k0 < K4; k0 += 4) {
    const int ka = k0 + (half << 1);            // lanes 0-15: k0,k0+1 ; 16-31: k0+2,k0+3
    v2f a = *(const v2f*)(Arow + ka);           // global_load_b64
    v2f b;
    b.x = Bcol[(size_t)ka * N];
    b.y = Bcol[(size_t)(ka + 1) * N];
    acc = __builtin_amdgcn_wmma_f32_16x16x4_f32(
        false, a, false, b, (short)0, acc, false, false);
  }
  if (k0 < K) {                                 // wave-uniform tail (K % 4 == 2 here)
    const int ka  = k0 + (half << 1);
    const int ka0 = ka     < K ? ka     : K - 1;
    const int ka1 = ka + 1 < K ? ka + 1 : K - 1;
    float a0 = Arow[ka0], a1 = Arow[ka1];       // unconditional clamped loads
    float b0 = Bcol[(size_t)ka0 * N], b1 = Bcol[(size_t)ka1 * N];
    v2f a, b;
    a.x = ka     < K ? a0 : 0.f;                // v_cndmask, no branch
    a.y = ka + 1 < K ? a1 : 0.f;
    b.x = ka     < K ? b0 : 0.f;
    b.y = ka + 1 < K ? b1 : 0.f;
    acc = __builtin_amdgcn_wmma_f32_16x16x4_f32(
        false, a, false, b, (short)0, acc, false, false);
  }

  // C/D 16x16 f32 layout: VGPR i, lanes 0-15 -> M=row0+i ; lanes 16-31 -> M=row0+i+8
  float bv = 0.f;
  if (bias) bv = bias[ncol_c];
  float* __restrict__ Crow = C + ncol_c;
  if (row0 + 16 <= M && col0 + 16 <= N) {       // wave-uniform interior fast path
#pragma unroll
    for (int i = 0; i < 8; ++i) {
      float v = acc[i] + bv;
      if (act == 1) v = lrelu(v, 0.01f);
      else if (act == 2) v = fmaxf(v, 0.f);
      Crow[(size_t)(row0 + i + (half << 3)) * N] = v;
    }
  } else {
#pragma unroll
    for (int i = 0; i < 8; ++i) {
      const int row = row0 + i + (half << 3);
      if (row < M && nvalid) {
        float v = acc[i] + bv;
        if (act == 1) v = lrelu(v, 0.01f);
        else if (act == 2) v = fmaxf(v, 0.f);
        Crow[(size_t)row * N] = v;
      }
    }
  }
}

// ---------------- utility kernels ----------------
__global__ void k_zero_f32(float* p, long long n) {
  long long i = (long long)blockIdx.x * blockDim.x + threadIdx.x;
  if (i < n) p[i] = 0.f;
}
__global__ void k_fill_u32(unsigned* p, long long n, unsigned v) {
  long long i = (long long)blockIdx.x * blockDim.x + threadIdx.x;
  if (i < n) p[i] = v;
}

// degree over dst (E edges + N self loops)
__global__ void k_deg(const int* __restrict__ ei, int E, int N, float* __restrict__ deg) {
  int i = blockIdx.x * blockDim.x + threadIdx.x;
  if (i < E)          atomicAdd(&deg[ei[E + i]], 1.f);
  else if (i < E + N) atomicAdd(&deg[i - E], 1.f);
}
__global__ void k_dinv(const float* __restrict__ deg, float* __restrict__ dinv, int N) {
  int i = blockIdx.x * blockDim.x + threadIdx.x;
  if (i < N) {
    float d = deg[i];
    dinv[i] = d > 0.f ? rsqrtf(fmaxf(d, 1.f)) : 0.f;
  }
}

// GCN scatter: h[dst] += h0[src] * dinv[src]*dinv[dst]   (wave per edge)
__global__ void k_scatter_gcn(const int* __restrict__ ei, int E, int N,
                              const float* __restrict__ dinv,
                              const float* __restrict__ h0, float* __restrict__ h) {
  int wid  = (blockIdx.x * blockDim.x + threadIdx.x) >> 5;
  int lane = threadIdx.x & 31;
  if (wid >= E + N) return;
  int s, d;
  if (wid < E) { s = ei[wid]; d = ei[E + wid]; } else { s = d = wid - E; }
  float norm = dinv[s] * dinv[d];
  const float* hs = h0 + (size_t)s * F_MID;
  float*       hd = h  + (size_t)d * F_MID;
  __builtin_prefetch(hs, 0, 0);
  for (int f = lane; f < F_MID; f += 32)
    atomicAdd(&hd[f], hs[f] * norm);
}

// in-place bias + leaky over [total = rows*F] with bias length F
__global__ void k_bias_leaky(float* __restrict__ buf, const float* __restrict__ bias,
                             long long total, int F, float slope) {
  long long i = (long long)blockIdx.x * blockDim.x + threadIdx.x;
  if (i < total) {
    int f = (int)(i % F);
    buf[i] = lrelu(buf[i] + bias[f], slope);
  }
}

// per (node, head) attention coefficients a_src / a_dst
__global__ void k_att_coef(const float* __restrict__ g0,
                           const float* __restrict__ att_src,
                           const float* __restrict__ att_dst,
                           float* __restrict__ as, float* __restrict__ ad, int N) {
  int i = blockIdx.x * blockDim.x + threadIdx.x;
  if (i >= N * 2) return;
  int n = i >> 1, hd = i & 1;
  const float* gp = g0 + (size_t)n * F_OUT + hd * F_MID;
  const float* ps = att_src + hd * F_MID;
  const float* pd = att_dst + hd * F_MID;
  float s0 = 0.f, s1 = 0.f;
  for (int f = 0; f < F_MID; ++f) { float g = gp[f]; s0 += g * ps[f]; s1 += g * pd[f]; }
  as[i] = s0; ad[i] = s1;
}

// per (edge, head): e = leaky(a_src[src]+a_dst[dst], 0.2); segment max via u32 atomicMax
__global__ void k_edge_logit_max(const int* __restrict__ ei, int E, int N,
                                 const float* __restrict__ as, const float* __restrict__ ad,
                                 float* __restrict__ ebuf, unsigned* __restrict__ mmax) {
  int i = blockIdx.x * blockDim.x + threadIdx.x;
  int id = i >> 1, hd = i & 1;
  if (id >= E + N) return;
  int s, d;
  if (id < E) { s = ei[id]; d = ei[E + id]; } else { s = d = id - E; }
  float e = as[s * 2 + hd] + ad[d * 2 + hd];
  e = lrelu(e, 0.2f);
  ebuf[i] = e;
  atomicMax(&mmax[d * 2 + hd], fmap_ord(e));
}

// per (edge, head): alpha = exp(e - m[dst]); denom[dst] += alpha (alpha overwrites ebuf)
__global__ void k_alpha(const int* __restrict__ ei, int E, int N,
                        const unsigned* __restrict__ mmax,
                        float* __restrict__ ebuf, float* __restrict__ denom) {
  int i = blockIdx.x * blockDim.x + threadIdx.x;
  int id = i >> 1, hd = i & 1;
  if (id >= E + N) return;
  int d = (id < E) ? ei[E + id] : (id - E);
  float a = expf(ebuf[i] - funmap_ord(mmax[d * 2 + hd]));
  ebuf[i] = a;
  atomicAdd(&denom[d * 2 + hd], a);
}

// GAT scatter: gout[dst] += g0[src] * alpha/denom   (wave per edge, 312 feats)
__global__ void k_scatter_gat(const int* __restrict__ ei, int E, int N,
                              const float* __restrict__ alpha,
                              const float* __restrict__ denom,
                              const float* __restrict__ g0, float* __restrict__ gout) {
  int wid  = (blockIdx.x * blockDim.x + threadIdx.x) >> 5;
  int lane = threadIdx.x & 31;
  if (wid >= E + N) return;
  int s, d;
  if (wid < E) { s = ei[wid]; d = ei[E + wid]; } else { s = d = wid - E; }
  float w0 = alpha[wid * 2 + 0] / denom[d * 2 + 0];
  float w1 = alpha[wid * 2 + 1] / denom[d * 2 + 1];
  const float* gs = g0   + (size_t)s * F_OUT;
  float*       gd = gout + (size_t)d * F_OUT;
  __builtin_prefetch(gs, 0, 0);
  for (int f = lane; f < F_OUT; f += 32)
    atomicAdd(&gd[f], gs[f] * (f < F_MID ? w0 : w1));
}

// global max pool: pu[batch[n]][f] = max(..., g[n][f]) via u32 map
__global__ void k_pool_max(const int* __restrict__ batch, const float* __restrict__ g,
                           unsigned* __restrict__ pu, int N) {
  long long i = (long long)blockIdx.x * blockDim.x + threadIdx.x;
  if (i >= (long long)N * F_OUT) return;
  int n = (int)(i / F_OUT), f = (int)(i % F_OUT);
  atomicMax(&pu[(size_t)batch[n] * F_OUT + f], fmap_ord(g[i]));
}
__global__ void k_pool_fin(const unsigned* __restrict__ pu, float* __restrict__ pf, int total) {
  int i = blockIdx.x * blockDim.x + threadIdx.x;
  if (i < total) pf[i] = funmap_ord(pu[i]);
}

// ---------------- host side ----------------
static inline void launch_gemm(const float* A, const float* B, const float* bias, float* C,
                               int M, int K, int N, int act, hipStream_t s) {
  int tiles = ((M + 15) / 16) * ((N + 15) / 16);
  int blocks = (tiles + 7) / 8;                 // 8 waves (256 threads) per block
  hipLaunchKernelGGL(wmma_gemm_f32, dim3(blocks), dim3(256), 0, s, A, B, bias, C, M, K, N, act);
}

extern "C" void kernel_launch(void* const* d_in, const int* in_sizes, int n_in,
                              void* d_out, int out_size, void* d_ws, size_t ws_size,
                              hipStream_t stream) {
  const float* x       = (const float*)d_in[0];
  const int*   ei      = (const int*)  d_in[1];   // [2,E] flattened: row0=src, row1=dst
  const int*   batch   = (const int*)  d_in[2];
  const float* W1      = (const float*)d_in[3];
  const float* b1      = (const float*)d_in[4];
  const float* W2      = (const float*)d_in[5];
  const float* att_src = (const float*)d_in[6];
  const float* att_dst = (const float*)d_in[7];
  const float* b2      = (const float*)d_in[8];
  const float* Wg      = (const float*)d_in[9];
  const float* bg      = (const float*)d_in[10];
  const float* Wf1     = (const float*)d_in[11];
  const float* bf1     = (const float*)d_in[12];
  const float* Wf2     = (const float*)d_in[13];
  const float* bf2     = (const float*)d_in[14];
  const float* Wo      = (const float*)d_in[15];
  const float* bo      = (const float*)d_in[16];
  float* out = (float*)d_out;

  const int N = in_sizes[0] / F_IN;    // 50000
  const int E = in_sizes[1] / 2;       // 800000
  const int EN = E + N;                // edges incl. self loops

  // ---- workspace carve-out (256B aligned) ----
  char* base = (char*)d_ws;
  size_t off = 0;
  auto carve = [&](size_t bytes) -> char* {
    char* p = base + off;
    off = (off + bytes + 255) & ~(size_t)255;
    return p;
  };
  float*    deg   = (float*)   carve((size_t)N * 4);
  float*    dinv  = (float*)   carve((size_t)N * 4);
  char*     h0c   =            carve((size_t)N * F_MID * 4);   // GCN pre-aggregate
  float*    h0    = (float*)h0c;
  float*    h     = (float*)   carve((size_t)N * F_MID * 4);   // GCN output
  float*    gout  = (float*)h0c;  // reuse: h0+h region (2*N*156*4 >= N*312*4); dead by then
  float*    g0    = (float*)   carve((size_t)N * F_OUT * 4);   // GAT pre-aggregate
  float*    as_   = (float*)   carve((size_t)N * 2 * 4);
  float*    ad_   = (float*)   carve((size_t)N * 2 * 4);
  float*    ebuf  = (float*)   carve((size_t)EN * 2 * 4);      // logits, then alphas
  unsigned* mmax  = (unsigned*)carve((size_t)N * 2 * 4);
  float*    denom = (float*)   carve((size_t)N * 2 * 4);
  unsigned* pu    = (unsigned*)carve((size_t)NB * F_OUT * 4);
  float*    pf    = (float*)   carve((size_t)NB * F_OUT * 4);
  float*    t1    = (float*)   carve((size_t)NB * 128 * 4);
  float*    t2    = (float*)   carve((size_t)NB * 1024 * 4);
  float*    t3    = (float*)   carve((size_t)NB * 512 * 4);
  (void)ws_size; (void)n_in; (void)out_size;

  const int T = 256;
  auto blk  = [](long long n, int t) { return (unsigned)((n + t - 1) / t); };

  // ---- GCNConv ----
  hipLaunchKernelGGL(k_zero_f32, dim3(blk(N, T)), dim3(T), 0, stream, deg, (long long)N);
  hipLaunchKernelGGL(k_deg,      dim3(blk(EN, T)), dim3(T), 0, stream, ei, E, N, deg);
  hipLaunchKernelGGL(k_dinv,     dim3(blk(N, T)), dim3(T), 0, stream, deg, dinv, N);
  launch_gemm(x, W1, nullptr, h0, N, F_IN, F_MID, 0, stream);                    // WMMA
  hipLaunchKernelGGL(k_zero_f32, dim3(blk((long long)N * F_MID, T)), dim3(T), 0, stream,
                     h, (long long)N * F_MID);
  hipLaunchKernelGGL(k_scatter_gcn, dim3(blk((long long)EN * 32, T)), dim3(T), 0, stream,
                     ei, E, N, dinv, h0, h);
  hipLaunchKernelGGL(k_bias_leaky, dim3(blk((long long)N * F_MID, T)), dim3(T), 0, stream,
                     h, b1, (long long)N * F_MID, F_MID, 0.01f);

  // ---- GATConv ----
  launch_gemm(h, W2, nullptr, g0, N, F_MID, F_OUT, 0, stream);                   // WMMA
  hipLaunchKernelGGL(k_att_coef, dim3(blk((long long)N * 2, T)), dim3(T), 0, stream,
                     g0, att_src, att_dst, as_, ad_, N);
  hipLaunchKernelGGL(k_fill_u32, dim3(blk((long long)N * 2, T)), dim3(T), 0, stream,
                     mmax, (long long)N * 2, NEG_INF_ORD);
  hipLaunchKernelGGL(k_edge_logit_max, dim3(blk((long long)EN * 2, T)), dim3(T), 0, stream,
                     ei, E, N, as_, ad_, ebuf, mmax);
  hipLaunchKernelGGL(k_zero_f32, dim3(blk((long long)N * 2, T)), dim3(T), 0, stream,
                     denom, (long long)N * 2);
  hipLaunchKernelGGL(k_alpha, dim3(blk((long long)EN * 2, T)), dim3(T), 0, stream,
                     ei, E, N, mmax, ebuf, denom);
  hipLaunchKernelGGL(k_zero_f32, dim3(blk((long long)N * F_OUT, T)), dim3(T), 0, stream,
                     gout, (long long)N * F_OUT);
  hipLaunchKernelGGL(k_scatter_gat, dim3(blk((long long)EN * 32, T)), dim3(T), 0, stream,
                     ei, E, N, ebuf, denom, g0, gout);
  hipLaunchKernelGGL(k_bias_leaky, dim3(blk((long long)N * F_OUT, T)), dim3(T), 0, stream,
                     gout, b2, (long long)N * F_OUT, F_OUT, 0.01f);

  // ---- global max pool ----
  hipLaunchKernelGGL(k_fill_u32, dim3(blk((long long)NB * F_OUT, T)), dim3(T), 0, stream,
                     pu, (long long)NB * F_OUT, NEG_INF_ORD);
  hipLaunchKernelGGL(k_pool_max, dim3(blk((long long)N * F_OUT, T)), dim3(T), 0, stream,
                     batch, gout, pu, N);
  hipLaunchKernelGGL(k_pool_fin, dim3(blk((long long)NB * F_OUT, T)), dim3(T), 0, stream,
                     pu, pf, NB * F_OUT);

  // ---- MLP head (all WMMA) ----
  launch_gemm(pf, Wg,  bg,  t1,  NB, F_OUT, 128,  1, stream);
  launch_gemm(t1, Wf1, bf1, t2,  NB, 128,   1024, 2, stream);
  launch_gemm(t2, Wf2, bf2, t3,  NB, 1024,  512,  2, stream);
  launch_gemm(t3, Wo,  bo,  out, NB, 512,   1,    0, stream);
}